// FFTConv_12025908428819
// MI455X (gfx1250) — compile-verified
//
#include <hip/hip_runtime.h>

// FFT tile convolution for MI455X (gfx1250): all heavy math as fp32 WMMA GEMMs,
// all global->LDS staging via CDNA5 async LDS loads (ASYNCcnt) when available.
//
// Pipeline (all kernels on `stream`, serialized by the stream):
//  init : build DFT matrices A1 (64x80, pad-aware forward DFT) and Wv (80x80 inverse)
//  T0   : kernel_real/imag [o][i][f] -> KF [f][o][i]        (tiled transpose)
//  A0   : x NHWC -> XTile [(m*64+i)][y][x]                  (tiled transpose)
//  A    : per (m,i): T1 = X@A1 ; T2 = A1^T@T1 (complex)  -> XT [(mi)][f]   (WMMA)
//  T1t  : XT [mi][f] -> XF [f][mi]                          (tiled transpose)
//  B    : per f: Y[m][o] = sum_i X[m][i]*K[i][o] (complex) -> YF [f][mo]   (WMMA)
//  T2t  : YF [f][mo] -> YT [mo][f]                          (tiled transpose)
//  C    : per (m,o): Z = Y@Wv ; out = Re(Wv@Z)           -> OT [mo][p][q]  (WMMA)
//  D    : overlap-add stitch + bias via LDS row accumulator -> out NHWC
//
// Workspace requirement: ~909 MB (3 double slabs of 8192*4624 floats + consts).

#define TN    4
#define TILE_ 64
#define NB    8
#define CIN   64
#define COUT  64
#define IMG_  256
#define FFTN  68
#define PADP  2
#define M_    128            // NB*TN*TN
#define F2    4624           // FFTN*FFTN
#define FP    80             // freq dim padded to multiple of 16
#define MI    8192           // M_*CIN
#define MO    8192           // M_*COUT

// LDS leading dims: multiples of 4 (16B-aligned rows for async b128) chosen for
// conflict-free WMMA fragment access: 68 -> banks 4*mr, 84 -> banks 20*mr.
#define LDB   68
#define LDC   84

typedef float v2f __attribute__((ext_vector_type(2)));
typedef float v8f __attribute__((ext_vector_type(8)));

// ---------------- async global->LDS copy (CDNA5 ASYNCcnt path) ----------------

#if __has_builtin(__builtin_amdgcn_global_load_async_to_lds_b128) && \
    __has_builtin(__builtin_amdgcn_s_wait_asynccnt)
#define HAS_ASYNC 1
#endif

// Exact parameter types per hipcc diagnostic: AS1 / AS3 pointers to int16-byte vectors.
typedef int v4i_t __attribute__((vector_size(4 * sizeof(int))));
typedef __attribute__((address_space(1))) v4i_t gv4i;
typedef __attribute__((address_space(3))) v4i_t lv4i;

__device__ __forceinline__ void cp16(float* l, const float* g) {
#ifdef HAS_ASYNC
  // Integer casts sidestep generic->AS pointer cast rules; low 32 bits of a
  // flat LDS address are the LDS byte offset (aperture rules).
  __builtin_amdgcn_global_load_async_to_lds_b128(
      (gv4i*)(unsigned long long)g,
      (lv4i*)(unsigned)(unsigned long long)l, 0, 0);
#else
  *(float4*)l = *(const float4*)g;
#endif
}
__device__ __forceinline__ void cp_join() {
#ifdef HAS_ASYNC
  __builtin_amdgcn_s_wait_asynccnt(0);
#endif
}

// ---------------- WMMA helpers (V_WMMA_F32_16X16X4_F32) ----------------

__device__ __forceinline__ v8f wmma4(v2f a, v2f b, v8f c) {
  return __builtin_amdgcn_wmma_f32_16x16x4_f32(false, a, false, b, (short)0, c,
                                               false, false);
}

// A fragment 16x4 at (r0,k0), row-major LDS, leading dim ld.
__device__ __forceinline__ v2f fragA(const float* s, int ld, int r0, int k0, int lane) {
  int h = lane >> 4, mr = lane & 15;
  v2f a;
  a.x = s[(r0 + mr) * ld + k0 + 2 * h];
  a.y = s[(r0 + mr) * ld + k0 + 2 * h + 1];
  return a;
}
// A fragment from transposed storage: A[m][k] = s[k*ld + m]
__device__ __forceinline__ v2f fragAT(const float* s, int ld, int r0, int k0, int lane) {
  int h = lane >> 4, mr = lane & 15;
  v2f a;
  a.x = s[(k0 + 2 * h) * ld + r0 + mr];
  a.y = s[(k0 + 2 * h + 1) * ld + r0 + mr];
  return a;
}
// B fragment 4x16 at (k0,c0), row-major LDS
__device__ __forceinline__ v2f fragB(const float* s, int ld, int k0, int c0, int lane) {
  int h = lane >> 4, nn = lane & 15;
  v2f b;
  b.x = s[(k0 + 2 * h) * ld + c0 + nn];
  b.y = s[(k0 + 2 * h + 1) * ld + c0 + nn];
  return b;
}
// B fragment from transposed storage: B[k][n] = s[n*ld + k]
__device__ __forceinline__ v2f fragBT(const float* s, int ld, int k0, int c0, int lane) {
  int h = lane >> 4, nn = lane & 15;
  v2f b;
  b.x = s[(c0 + nn) * ld + k0 + 2 * h];
  b.y = s[(c0 + nn) * ld + k0 + 2 * h + 1];
  return b;
}
// C/D fragment store
__device__ __forceinline__ void storeC(float* s, int ld, int r0, int c0, int lane, v8f c) {
  int h = lane >> 4, nn = lane & 15;
#pragma unroll
  for (int v = 0; v < 8; ++v) s[(r0 + v + 8 * h) * ld + c0 + nn] = c[v];
}

// ---------------- init: DFT matrices ----------------

__global__ void k_init_dft(float* a1r, float* a1i, float* wvr, float* wvi) {
  const float twopi = 6.283185307179586f;
  for (int idx = threadIdx.x; idx < 64 * FP; idx += blockDim.x) {
    int n = idx / FP, k = idx % FP;
    float vr = 0.f, vi = 0.f;
    if (k < FFTN) {
      int t = (k * (n + PADP)) % FFTN;
      float th = twopi * (float)t / (float)FFTN;
      vr = cosf(th);
      vi = -sinf(th);
    }
    a1r[idx] = vr;
    a1i[idx] = vi;
  }
  for (int idx = threadIdx.x; idx < FP * FP; idx += blockDim.x) {
    int p = idx / FP, q = idx % FP;
    float vr = 0.f, vi = 0.f;
    if (p < FFTN && q < FFTN) {
      int t = (p * q) % FFTN;
      float th = twopi * (float)t / (float)FFTN;
      vr = cosf(th) / (float)FFTN;
      vi = sinf(th) / (float)FFTN;
    }
    wvr[idx] = vr;
    wvi[idx] = vi;
  }
}

// ---------------- tiled transpose (pair of arrays via blockIdx.z) ----------------
// All our shapes have R,C multiples of 16, so the 4-wide groups never straddle C.

__global__ void k_transpose2(const float* __restrict__ ir, const float* __restrict__ ii,
                             float* __restrict__ outr, float* __restrict__ outi,
                             int R, int C) {
  __shared__ float t[64][LDB];
  const float* in = blockIdx.z ? ii : ir;
  float* out = blockIdx.z ? outi : outr;
  int r0 = blockIdx.y * 64, c0 = blockIdx.x * 64;
  int lr = threadIdx.x >> 4;          // 0..15
  int lc = (threadIdx.x & 15) * 4;    // 0,4,..,60
#pragma unroll
  for (int k = 0; k < 4; ++k) {
    int r = r0 + lr + k * 16, c = c0 + lc;
    if (r < R && c + 3 < C) cp16(&t[lr + k * 16][lc], &in[(size_t)r * C + c]);
  }
  cp_join();
  __syncthreads();
  int tr = threadIdx.x >> 6, tc = threadIdx.x & 63;
#pragma unroll
  for (int k = 0; k < 16; ++k) {
    int c = c0 + tr + k * 4, r = r0 + tc;
    if (c < C && r < R) out[(size_t)c * R + r] = t[tc][tr + k * 4];
  }
}

// ---------------- A0: x NHWC -> XTile[(m*64+i)][y][x] ----------------

__global__ void k_pack_x(const float* __restrict__ x, float* __restrict__ xt) {
  __shared__ float t[64][LDB];
  int m = blockIdx.x, y = blockIdx.y;
  int b = m >> 4, r = (m >> 2) & 3, c = m & 3;
  int gy = r * TILE_ + y;
  size_t src = ((size_t)(b * IMG_ + gy) * IMG_ + c * TILE_) * CIN;
  int lr = threadIdx.x >> 4, lc = (threadIdx.x & 15) * 4;
#pragma unroll
  for (int k = 0; k < 4; ++k)
    cp16(&t[lr + k * 16][lc], &x[src + (size_t)(lr + k * 16) * CIN + lc]);
  cp_join();
  __syncthreads();
  for (int idx = threadIdx.x; idx < 4096; idx += blockDim.x) {
    int i = idx >> 6, xx = idx & 63;
    xt[(((size_t)(m * 64 + i)) * 64 + y) * 64 + xx] = t[xx][i];
  }
}

// ---------------- A: forward DFT per (m,i) via WMMA ----------------

__global__ void k_fft_fwd(const float* __restrict__ xt,
                          const float* __restrict__ a1r, const float* __restrict__ a1i,
                          float* __restrict__ xfr, float* __restrict__ xfi) {
  extern __shared__ float sm[];
  float* sX  = sm;                 // 64*LDB
  float* sAr = sX + 64 * LDB;      // 64*80
  float* sAi = sAr + 64 * FP;
  float* sTr = sAi + 64 * FP;      // 64*80
  float* sTi = sTr + 64 * FP;
  int g = blockIdx.x;              // mi = m*64 + i
  int tid = threadIdx.x, lane = tid & 31, wave = tid >> 5;

  for (int idx = tid * 4; idx < 4096; idx += blockDim.x * 4)
    cp16(&sX[(idx >> 6) * LDB + (idx & 63)], &xt[(size_t)g * 4096 + idx]);
  for (int idx = tid * 4; idx < 64 * FP; idx += blockDim.x * 4) {
    cp16(&sAr[idx], &a1r[idx]);
    cp16(&sAi[idx], &a1i[idx]);
  }
  cp_join();
  __syncthreads();

  // Step 1: T1 = X @ A1   (real x complex): M=64, K=64, N=80
  for (int t = wave; t < 20; t += 8) {
    int mt = t / 5, nt = t % 5;
    v8f cr = {}, ci = {};
    for (int k = 0; k < 64; k += 4) {
      v2f a  = fragA(sX, LDB, mt * 16, k, lane);
      v2f br = fragB(sAr, FP, k, nt * 16, lane);
      v2f bi = fragB(sAi, FP, k, nt * 16, lane);
      cr = wmma4(a, br, cr);
      ci = wmma4(a, bi, ci);
    }
    storeC(sTr, FP, mt * 16, nt * 16, lane, cr);
    storeC(sTi, FP, mt * 16, nt * 16, lane, ci);
  }
  __syncthreads();

  // Step 2: T2 = A1^T @ T1 (complex x complex): M=80, K=64, N=80
  for (int t = wave; t < 25; t += 8) {
    int mt = t / 5, nt = t % 5;
    v8f P = {}, Q = {}, R = {}, S = {};
    for (int k = 0; k < 64; k += 4) {
      v2f ar = fragAT(sAr, FP, mt * 16, k, lane);  // A2[k1][n] = A1[n][k1]
      v2f ai = fragAT(sAi, FP, mt * 16, k, lane);
      v2f br = fragB(sTr, FP, k, nt * 16, lane);
      v2f bi = fragB(sTi, FP, k, nt * 16, lane);
      P = wmma4(ar, br, P);
      Q = wmma4(ai, bi, Q);
      R = wmma4(ar, bi, R);
      S = wmma4(ai, br, S);
    }
    v8f t2r = P - Q, t2i = R + S;
    int h = lane >> 4, nn = lane & 15;
    int k2 = nt * 16 + nn;
    if (k2 < FFTN) {
#pragma unroll
      for (int v = 0; v < 8; ++v) {
        int k1 = mt * 16 + v + 8 * h;
        if (k1 < FFTN) {
          size_t o = (size_t)g * F2 + k1 * FFTN + k2;
          xfr[o] = t2r[v];
          xfi[o] = t2i[v];
        }
      }
    }
  }
}

// ---------------- B: per-frequency complex GEMM [128x64]x[64x64] ----------------

__global__ void k_freq_gemm(const float* __restrict__ xfr, const float* __restrict__ xfi,
                            const float* __restrict__ kfr, const float* __restrict__ kfi,
                            float* __restrict__ yfr, float* __restrict__ yfi) {
  extern __shared__ float sm[];
  float* sXr = sm;                    // 128*LDB
  float* sXi = sXr + 128 * LDB;
  float* sKr = sXi + 128 * LDB;       // 64*LDB  (stored [o][i])
  float* sKi = sKr + 64 * LDB;
  int f = blockIdx.x;
  int tid = threadIdx.x, lane = tid & 31, wave = tid >> 5;

  for (int idx = tid * 4; idx < MI; idx += blockDim.x * 4) {
    int mm = idx >> 6, i = idx & 63;
    cp16(&sXr[mm * LDB + i], &xfr[(size_t)f * MI + idx]);
    cp16(&sXi[mm * LDB + i], &xfi[(size_t)f * MI + idx]);
  }
  for (int idx = tid * 4; idx < 4096; idx += blockDim.x * 4) {
    int o = idx >> 6, i = idx & 63;
    cp16(&sKr[o * LDB + i], &kfr[(size_t)f * 4096 + idx]);
    cp16(&sKi[o * LDB + i], &kfi[(size_t)f * 4096 + idx]);
  }
  cp_join();
  __syncthreads();

  int mt = wave;  // 8 waves cover M=128
  // Hoist all A fragments for this m-tile: reused across the 4 n-tiles.
  v2f aR[16], aI[16];
#pragma unroll
  for (int k = 0; k < 16; ++k) {
    aR[k] = fragA(sXr, LDB, mt * 16, k * 4, lane);
    aI[k] = fragA(sXi, LDB, mt * 16, k * 4, lane);
  }
  for (int nt = 0; nt < 4; ++nt) {
    v8f P = {}, Q = {}, R = {}, S = {};
#pragma unroll
    for (int k = 0; k < 16; ++k) {
      v2f br = fragBT(sKr, LDB, k * 4, nt * 16, lane);  // B[i][o] = sK[o][i]
      v2f bi = fragBT(sKi, LDB, k * 4, nt * 16, lane);
      P = wmma4(aR[k], br, P);
      Q = wmma4(aI[k], bi, Q);
      R = wmma4(aR[k], bi, R);
      S = wmma4(aI[k], br, S);
    }
    v8f yr = P - Q, yi = R + S;
    int h = lane >> 4, nn = lane & 15;
#pragma unroll
    for (int v = 0; v < 8; ++v) {
      int mm = mt * 16 + v + 8 * h;
      size_t o = (size_t)f * MO + (size_t)mm * 64 + nt * 16 + nn;
      yfr[o] = yr[v];
      yfi[o] = yi[v];
    }
  }
}

// ---------------- C: inverse DFT per (m,o), real output ----------------

__global__ void k_fft_inv(const float* __restrict__ ytr, const float* __restrict__ yti,
                          const float* __restrict__ wvr, const float* __restrict__ wvi,
                          float* __restrict__ ot) {
  extern __shared__ float sm[];
  float* sYr = sm;                  // 80*LDC
  float* sYi = sYr + FP * LDC;
  float* sWr = sYi + FP * LDC;
  float* sWi = sWr + FP * LDC;
  float* sZr = sWi + FP * LDC;
  float* sZi = sZr + FP * LDC;
  int g = blockIdx.x;  // mo = m*64 + o
  int tid = threadIdx.x, lane = tid & 31, wave = tid >> 5;

  for (int idx = tid; idx < FP * LDC; idx += blockDim.x) {
    sYr[idx] = 0.f;
    sYi[idx] = 0.f;
  }
  __syncthreads();  // LDS zero-fill visible before async writes land
  for (int idx = tid * 4; idx < F2; idx += blockDim.x * 4) {
    int p = idx / FFTN, q = idx % FFTN;  // q stays a multiple of 4 (68%4==0)
    cp16(&sYr[p * LDC + q], &ytr[(size_t)g * F2 + idx]);
    cp16(&sYi[p * LDC + q], &yti[(size_t)g * F2 + idx]);
  }
  for (int idx = tid * 4; idx < FP * FP; idx += blockDim.x * 4) {
    int p = idx / FP, q = idx % FP;
    cp16(&sWr[p * LDC + q], &wvr[idx]);
    cp16(&sWi[p * LDC + q], &wvi[idx]);
  }
  cp_join();
  __syncthreads();

  // Step 1: Z = Y @ Wv : M=80, K=68, N=80
  for (int t = wave; t < 25; t += 8) {
    int mt = t / 5, nt = t % 5;
    v8f P = {}, Q = {}, R = {}, S = {};
    for (int k = 0; k < FFTN; k += 4) {
      v2f ar = fragA(sYr, LDC, mt * 16, k, lane);
      v2f ai = fragA(sYi, LDC, mt * 16, k, lane);
      v2f br = fragB(sWr, LDC, k, nt * 16, lane);
      v2f bi = fragB(sWi, LDC, k, nt * 16, lane);
      P = wmma4(ar, br, P);
      Q = wmma4(ai, bi, Q);
      R = wmma4(ar, bi, R);
      S = wmma4(ai, br, S);
    }
    storeC(sZr, LDC, mt * 16, nt * 16, lane, P - Q);
    storeC(sZi, LDC, mt * 16, nt * 16, lane, R + S);
  }
  __syncthreads();

  // Step 2: out = Re(Wv @ Z) : M=80, K=68, N=80 (Wv symmetric)
  for (int t = wave; t < 25; t += 8) {
    int mt = t / 5, nt = t % 5;
    v8f P = {}, Q = {};
    for (int k = 0; k < FFTN; k += 4) {
      v2f ar = fragA(sWr, LDC, mt * 16, k, lane);
      v2f ai = fragA(sWi, LDC, mt * 16, k, lane);
      v2f br = fragB(sZr, LDC, k, nt * 16, lane);
      v2f bi = fragB(sZi, LDC, k, nt * 16, lane);
      P = wmma4(ar, br, P);
      Q = wmma4(ai, bi, Q);
    }
    v8f res = P - Q;
    int h = lane >> 4, nn = lane & 15;
    int qq = nt * 16 + nn;
    if (qq < FFTN) {
#pragma unroll
      for (int v = 0; v < 8; ++v) {
        int pp = mt * 16 + v + 8 * h;
        if (pp < FFTN) ot[(size_t)g * F2 + pp * FFTN + qq] = res[v];
      }
    }
  }
}

// ---------------- D: overlap-add stitch + bias, race-free via LDS row acc ----------------

__global__ void k_stitch(const float* __restrict__ ot, const float* __restrict__ bias,
                         float* __restrict__ out) {
  extern __shared__ float acc[];  // IMG_*64 floats = 64 KB
  int y = blockIdx.x, b = blockIdx.y;
  int tid = threadIdx.x;
  for (int idx = tid; idx < IMG_ * 64; idx += blockDim.x) acc[idx] = bias[idx & 63];
  __syncthreads();
  for (int rv = 0; rv < TN; ++rv) {
    int p = y - rv * (FFTN - PADP);
    if (p < 0 || p >= FFTN) continue;
    for (int c = 0; c < TN; ++c) {
      int m = b * 16 + rv * 4 + c;
      for (int t = tid; t < 64 * FFTN; t += blockDim.x) {
        int o = t / FFTN, q = t - o * FFTN;
        int xg = c * (FFTN - PADP) + q;
        if (xg < IMG_)
          acc[xg * 64 + o] += ot[((size_t)(m * 64 + o)) * F2 + p * FFTN + q];
      }
      __syncthreads();
    }
  }
  size_t base = ((size_t)(b * IMG_ + y)) * IMG_ * 64;
  for (int idx = tid; idx < IMG_ * 64; idx += blockDim.x) out[base + idx] = acc[idx];
}

// ---------------- host launcher ----------------

extern "C" void kernel_launch(void* const* d_in, const int* in_sizes, int n_in,
                              void* d_out, int out_size, void* d_ws, size_t ws_size,
                              hipStream_t stream) {
  (void)in_sizes; (void)n_in; (void)out_size; (void)ws_size;
  const float* x    = (const float*)d_in[0];
  const float* krr  = (const float*)d_in[1];
  const float* kri  = (const float*)d_in[2];
  const float* bias = (const float*)d_in[3];
  float* out = (float*)d_out;

  char* w = (char*)d_ws;
  size_t off = 0;
  auto alloc = [&](size_t nfloats) -> float* {
    float* p = (float*)(w + off);
    off += ((nfloats * sizeof(float) + 255) / 256) * 256;
    return p;
  };
  float* a1r = alloc(64 * FP);
  float* a1i = alloc(64 * FP);
  float* wvr = alloc(FP * FP);
  float* wvi = alloc(FP * FP);
  const size_t slab = (size_t)MI * F2;  // 37,879,808 floats
  float* s1r = alloc(slab);  // XT  -> later YF
  float* s1i = alloc(slab);
  float* s2r = alloc(slab);  // XTile, then XF -> later YT
  float* s2i = alloc(slab);
  float* s3r = alloc(slab);  // KF -> later OT (real)
  float* s3i = alloc(slab);

  dim3 blk(256);

  k_init_dft<<<dim3(1), blk, 0, stream>>>(a1r, a1i, wvr, wvi);

  // T0: kernel [oi=4096][f=4624] -> KF [f][oi]
  k_transpose2<<<dim3((F2 + 63) / 64, 4096 / 64, 2), blk, 0, stream>>>(
      krr, kri, s3r, s3i, 4096, F2);

  // A0: x -> XTile (in s2r)
  k_pack_x<<<dim3(M_, TILE_), blk, 0, stream>>>(x, s2r);

  // A: forward DFT -> XT (s1)
  size_t smA = (size_t)(64 * LDB + 4 * 64 * FP) * sizeof(float);
  k_fft_fwd<<<dim3(MI), blk, smA, stream>>>(s2r, a1r, a1i, s1r, s1i);

  // T1: XT [mi=8192][f=4624] -> XF [f][mi]
  k_transpose2<<<dim3((F2 + 63) / 64, MI / 64, 2), blk, 0, stream>>>(
      s1r, s1i, s2r, s2i, MI, F2);

  // B: per-frequency complex GEMM -> YF (s1; XT dead)
  size_t smB = (size_t)(2 * 128 * LDB + 2 * 64 * LDB) * sizeof(float);
  k_freq_gemm<<<dim3(F2), blk, smB, stream>>>(s2r, s2i, s3r, s3i, s1r, s1i);

  // T2: YF [f=4624][mo=8192] -> YT [mo][f] (s2; XF dead)
  k_transpose2<<<dim3((MI + 63) / 64, (F2 + 63) / 64, 2), blk, 0, stream>>>(
      s1r, s1i, s2r, s2i, F2, MI);

  // C: inverse DFT -> OT (s3r; KF dead)
  size_t smC = (size_t)(6 * FP * LDC) * sizeof(float);
  k_fft_inv<<<dim3(MO), blk, smC, stream>>>(s2r, s2i, wvr, wvi, s3r);

  // D: stitch + bias -> out
  size_t smD = (size_t)(IMG_ * 64) * sizeof(float);
  k_stitch<<<dim3(IMG_, NB), blk, smD, stream>>>(s3r, bias, out);
}